// CircleLossV2_21328807592494
// MI455X (gfx1250) — compile-verified
//
#include <hip/hip_runtime.h>
#include <hip/hip_bf16.h>
#include <math.h>

#define B_ROWS   8192
#define DIM      128
#define MARGIN   0.25f
#define GAMMA    256.0f
#define NEG_FILL (-1e9f)

typedef __attribute__((ext_vector_type(2))) float v2f;
typedef __attribute__((ext_vector_type(8))) float v8f;

// Online logsumexp update with a single v_exp_f32:
//   m' = max(m,l);  s' = s*exp(m-m') + exp(l-m')   (one factor is exp(0)=1)
__device__ __forceinline__ void lse_update(float& m, float& s, float l) {
    float d  = l - m;
    float t  = __expf(-fabsf(d));
    bool  gt = d > 0.0f;
    s = s * (gt ? t : 1.0f) + (gt ? 1.0f : t);
    m = gt ? l : m;
}

// Merge two (m,s) logsumexp states with a single v_exp_f32.
__device__ __forceinline__ void lse_merge(float& m, float& s, float m2, float s2) {
    float d  = m2 - m;
    float t  = __expf(-fabsf(d));
    bool  gt = d > 0.0f;
    s = gt ? (s * t + s2) : (s2 * t + s);
    m = gt ? m2 : m;
}

// ---------------------------------------------------------------------------
// Kernel 1: L2-normalize rows. One wave32 per row (4 floats per lane).
// ---------------------------------------------------------------------------
__global__ __launch_bounds__(256) void cl_normalize(const float* __restrict__ emb,
                                                    float* __restrict__ e) {
    int row  = blockIdx.x * 8 + (threadIdx.x >> 5);
    int lane = threadIdx.x & 31;
    const float4* src = (const float4*)(emb + row * DIM);
    float4 v = src[lane];
    float ss = v.x * v.x + v.y * v.y + v.z * v.z + v.w * v.w;
    #pragma unroll
    for (int m = 16; m > 0; m >>= 1) ss += __shfl_xor(ss, m, 32);
    float inv = 1.0f / fmaxf(sqrtf(ss), 1e-12f);
    float4 o;
    o.x = v.x * inv; o.y = v.y * inv; o.z = v.z * inv; o.w = v.w * inv;
    ((float4*)(e + row * DIM))[lane] = o;
}

// ---------------------------------------------------------------------------
// Kernel 2: fused WMMA sim-tile + circle-loss streaming logsumexp.
// 64 blocks x 256 threads (8 waves). Wave w handles a 16-row strip.
// Column tiles staged through padded LDS (stride 132 -> conflict-free b64
// reads); 4 independent WMMA accumulator chains hide XDL RAW latency.
// ---------------------------------------------------------------------------
#define BSTRIDE 132

__global__ __launch_bounds__(256) void cl_main(const float* __restrict__ e,
                                               const int* __restrict__ labels,
                                               float* __restrict__ row_val,
                                               float* __restrict__ row_valid) {
    __shared__ float btile[16 * BSTRIDE];   // 8448 B column tile (padded)
    __shared__ int   lab_s[B_ROWS];         // 32 KB: all labels LDS-resident

    const int tid  = threadIdx.x;
    const int wave = tid >> 5;
    const int lane = tid & 31;
    const int half = lane >> 4;   // 0: lanes 0-15, 1: lanes 16-31
    const int ln16 = lane & 15;

    const int row_base = blockIdx.x * 128 + wave * 16;

    for (int i = tid; i < B_ROWS; i += 256) lab_s[i] = labels[i];

    // Preload A fragments (16x4 f32 layout: lane L holds A[L%16][k0+2*(L>>4)+{0,1}])
    float2 afrag[32];
    const float* arow = e + (size_t)(row_base + ln16) * DIM + 2 * half;
    #pragma unroll
    for (int kk = 0; kk < 32; ++kk)
        afrag[kk] = *(const float2*)(arow + 4 * kk);

    __syncthreads();

    int lrow[8];
    #pragma unroll
    for (int v = 0; v < 8; ++v) lrow[v] = lab_s[row_base + v + 8 * half];

    float mp[8], sp[8], mq[8], sq[8];
    #pragma unroll
    for (int v = 0; v < 8; ++v) { mp[v] = -1e30f; sp[v] = 0.0f; mq[v] = -1e30f; sq[v] = 0.0f; }

    for (int ct = 0; ct < B_ROWS / 16; ++ct) {
        const int col_base = ct * 16;

        __syncthreads();  // previous tile's B reads complete
        #pragma unroll
        for (int t = 0; t < 2; ++t) {
            int f  = tid + t * 256;
            int r  = f >> 5;
            int c4 = f & 31;
            float4 vv = *(const float4*)(e + (size_t)(col_base + r) * DIM + c4 * 4);
            *(float4*)(&btile[r * BSTRIDE + c4 * 4]) = vv;
        }
        __syncthreads();

        // sim tile: 32 f32 WMMAs over K=128, 4 independent accumulator chains
        v8f acc[4] = {{}, {}, {}, {}};
        const float* brow = &btile[ln16 * BSTRIDE + 2 * half];
        #pragma unroll
        for (int kk = 0; kk < 32; ++kk) {
            v2f a; a.x = afrag[kk].x; a.y = afrag[kk].y;
            v2f b = *(const v2f*)(brow + 4 * kk);
            acc[kk & 3] = __builtin_amdgcn_wmma_f32_16x16x4_f32(
                false, a, false, b, (short)0, acc[kk & 3], false, false);
        }
        v8f c = (acc[0] + acc[1]) + (acc[2] + acc[3]);

        // Epilogue: masked logits + single-exp online LSE updates
        const int colg = col_base + ln16;
        const int lcol = lab_s[colg];
        #pragma unroll
        for (int v = 0; v < 8; ++v) {
            float s    = c[v];
            int   rowg = row_base + v + 8 * half;
            bool  eq   = (lrow[v] == lcol);
            bool  diag = (rowg == colg);

            float lp = (eq && !diag)
                         ? (-GAMMA * fmaxf(1.0f + MARGIN - s, 0.0f) * (s - (1.0f - MARGIN)))
                         : NEG_FILL;
            float ln = (!eq)
                         ? (GAMMA * fmaxf(s + MARGIN, 0.0f) * (s - MARGIN))
                         : NEG_FILL;

            lse_update(mp[v], sp[v], lp);
            lse_update(mq[v], sq[v], ln);
        }
    }

    // Merge per-lane partials across the 16-lane N group
    #pragma unroll
    for (int v = 0; v < 8; ++v) {
        #pragma unroll
        for (int m = 1; m <= 8; m <<= 1) {
            lse_merge(mp[v], sp[v], __shfl_xor(mp[v], m, 32), __shfl_xor(sp[v], m, 32));
            lse_merge(mq[v], sq[v], __shfl_xor(mq[v], m, 32), __shfl_xor(sq[v], m, 32));
        }
    }

    if (ln16 == 0) {  // lane 0 -> rows +0..7, lane 16 -> rows +8..15
        #pragma unroll
        for (int v = 0; v < 8; ++v) {
            int   r  = row_base + v + 8 * half;
            float lp = mp[v] + __logf(sp[v]);
            float ln = mq[v] + __logf(sq[v]);
            bool  valid = (lp > -1e8f) && (ln > -1e8f);  // real logits bounded ~1.3e3
            float z = lp + ln;
            float spv;
            if (z > 20.0f)       spv = z;
            else if (z < -20.0f) spv = __expf(z);
            else                 spv = log1pf(__expf(z));
            row_val[r]   = valid ? spv : 0.0f;
            row_valid[r] = valid ? 1.0f : 0.0f;
        }
    }
}

// ---------------------------------------------------------------------------
// Kernel 3: deterministic single-block reduction -> scalar loss
// ---------------------------------------------------------------------------
__global__ __launch_bounds__(256) void cl_finalize(const float* __restrict__ row_val,
                                                   const float* __restrict__ row_valid,
                                                   float* __restrict__ out) {
    __shared__ float ssum[256];
    __shared__ float scnt[256];
    float s = 0.0f, cn = 0.0f;
    for (int i = threadIdx.x; i < B_ROWS; i += 256) {
        s  += row_val[i];
        cn += row_valid[i];
    }
    ssum[threadIdx.x] = s;
    scnt[threadIdx.x] = cn;
    __syncthreads();
    for (int st = 128; st > 0; st >>= 1) {
        if (threadIdx.x < st) {
            ssum[threadIdx.x] += ssum[threadIdx.x + st];
            scnt[threadIdx.x] += scnt[threadIdx.x + st];
        }
        __syncthreads();
    }
    if (threadIdx.x == 0) out[0] = ssum[0] / fmaxf(scnt[0], 1.0f);
}

// ---------------------------------------------------------------------------
extern "C" void kernel_launch(void* const* d_in, const int* in_sizes, int n_in,
                              void* d_out, int out_size, void* d_ws, size_t ws_size,
                              hipStream_t stream) {
    (void)in_sizes; (void)n_in; (void)out_size; (void)ws_size;
    const float* emb    = (const float*)d_in[0];
    const int*   labels = (const int*)d_in[1];

    float* ws        = (float*)d_ws;
    float* e_norm    = ws;                          // 8192*128 floats (4 MB)
    float* row_val   = ws + (size_t)B_ROWS * DIM;   // 8192 floats
    float* row_valid = row_val + B_ROWS;            // 8192 floats

    cl_normalize<<<B_ROWS / 8, 256, 0, stream>>>(emb, e_norm);
    cl_main<<<B_ROWS / 128, 256, 0, stream>>>(e_norm, labels, row_val, row_valid);
    cl_finalize<<<1, 256, 0, stream>>>(row_val, row_valid, (float*)d_out);
}